// ROCStar_5806795784271
// MI455X (gfx1250) — compile-verified
//
#include <hip/hip_runtime.h>
#include <hip/hip_bf16.h>

// Problem constants (match reference)
#define BB 256
#define EE 8192
#define CC 19
#define MAXPOS 1000.0f

typedef __attribute__((ext_vector_type(2))) float v2f;
typedef __attribute__((ext_vector_type(8))) float v8f;

// relu as a single v_med3_num_f32: clamp(x, 0, 4).
// Valid because unmasked pair values satisfy d < 2.2 (< 4), and masked
// sentinel values (~ -1e7) clamp to 0. Both 0.0 and 4.0 are GCN inline
// constants, and the builtin lowers without a canonicalize.
__device__ __forceinline__ float relu_clamp(float x) {
  return __builtin_amdgcn_fmed3f(x, 0.0f, 4.0f);
}

// ---------------------------------------------------------------------------
// Kernel 1: per-class stats. One block per class, 256 threads.
//   stats[c*4+0] = p = 1000 / max(count(epoch_true>=0.5), 1)
//   stats[c*4+1] = degenerate flag (n_pos==0 || n_pos==B)
//   stats[c*4+2] = sum(y_pred[:,c])
// Deterministic tree reductions (no float atomics).
// ---------------------------------------------------------------------------
__global__ void rocstar_stats_kernel(const float* __restrict__ y_pred,
                                     const float* __restrict__ y_true,
                                     const float* __restrict__ epoch_true,
                                     float* __restrict__ stats) {
  const int c = blockIdx.x;
  const int t = threadIdx.x;  // 0..255

  __shared__ float s_sum[BB];
  __shared__ int   s_pos[BB];
  __shared__ int   s_cap[BB];

  const float yp = y_pred[t * CC + c];
  const float yt = y_true[t * CC + c];
  s_sum[t] = yp;
  s_pos[t] = (yt >= 0.5f) ? 1 : 0;

  int capc = 0;
  for (int k = 0; k < EE / BB; ++k) {
    const int j = t + BB * k;
    capc += (epoch_true[j * CC + c] >= 0.5f) ? 1 : 0;
  }
  s_cap[t] = capc;
  __syncthreads();

  for (int s = BB / 2; s > 0; s >>= 1) {
    if (t < s) {
      s_sum[t] += s_sum[t + s];
      s_pos[t] += s_pos[t + s];
      s_cap[t] += s_cap[t + s];
    }
    __syncthreads();
  }
  if (t == 0) {
    const int   npos = s_pos[0];
    const float cap  = fmaxf((float)s_cap[0], 1.0f);
    stats[c * 4 + 0] = MAXPOS / cap;
    stats[c * 4 + 1] = (npos == 0 || npos == BB) ? 1.0f : 0.0f;
    stats[c * 4 + 2] = s_sum[0];
  }
}

// ---------------------------------------------------------------------------
// Kernel 2: masked pairwise hinge^2 sums via WMMA difference matrices.
// grid = (64 j-blocks, 19 classes), block = 256 threads = 8 waves (wave32).
// Each wave owns 16 epoch columns; loops over all 16 batch i-tiles.
//
// D2[i][j] = (g - y_i) + e_j     (m2 pair value), masks folded in via -BIG
// D3[i][j] = (g + y_i) + (-e_j)  (m3 pair value), masks folded in via -BIG
//
// WMMA trick (layout-robust):  D = A x B + C with
//   A = a_i/4 replicated over all K (only depends on M = lane%16)
//   B = all ones                   (fully layout-independent)
//   C = b_j broadcast to all 8 acc VGPRs (C layout: N = lane%16, documented)
// => D[i][j] = a_i + b_j in full f32 precision, all pairwise sums on the
// matrix pipe; epilogue per element is one v_med3 + half a v_pk_fma.
// ---------------------------------------------------------------------------
__global__ void rocstar_pair_kernel(const float* __restrict__ y_pred,
                                    const float* __restrict__ y_true,
                                    const float* __restrict__ epoch_pred,
                                    const float* __restrict__ epoch_true,
                                    const float* __restrict__ gamma,
                                    const float* __restrict__ rand_pos,
                                    const float* __restrict__ rand_neg,
                                    const float* __restrict__ stats,
                                    float* __restrict__ partials) {
  const int c    = blockIdx.y;
  const int jb   = blockIdx.x;        // 0..63
  const int tid  = threadIdx.x;
  const int wave = tid >> 5;          // 0..7
  const int lane = tid & 31;
  const int l16  = lane & 15;

  const float g   = gamma[c];
  const float p   = stats[c * 4 + 0];
  const float BIG = 1.0e7f;

  // This wave's 16 epoch columns (lanes 16..31 mirror lanes 0..15)
  const int jj = jb * 128 + wave * 16 + l16;
  const float e  = epoch_pred[jj * CC + c];
  const float et = epoch_true[jj * CC + c];
  const float rp = rand_pos [jj * CC + c];
  const float rn = rand_neg [jj * CC + c];

  // Column masks folded into operand values:
  const float b2 = ((et <  0.5f) && (rn < p)) ?  e  : -BIG;  // epoch-neg subsample
  const float b3 = ((et >= 0.5f) && (rp < p)) ? (-e) : -BIG; // epoch-pos subsample

  v8f c2, c3;
#pragma unroll
  for (int r = 0; r < 8; ++r) { c2[r] = b2; c3[r] = b3; }
  v2f ones; ones[0] = 1.0f; ones[1] = 1.0f;

  // 4 independent accumulators to break the fmac RAW chain (fixed order);
  // adjacent pairs let the backend form v_pk_fma_f32.
  float acc0 = 0.0f, acc1 = 0.0f, acc2a = 0.0f, acc3a = 0.0f;

#pragma unroll 4
  for (int it = 0; it < 16; ++it) {
    const int i = it * 16 + l16;
    const float y   = y_pred[i * CC + c];
    const float yt  = y_true[i * CC + c];
    const bool  pos = (yt >= 0.5f);
    // Row masks folded into operand values:
    const float a2 = pos ? (g - y) : -BIG;   // batch positives for m2
    const float a3 = pos ? -BIG : (g + y);   // batch negatives for m3
    v2f A2; A2[0] = a2 * 0.25f; A2[1] = a2 * 0.25f;
    v2f A3; A3[0] = a3 * 0.25f; A3[1] = a3 * 0.25f;

    // (neg_a, A, neg_b, B, c_mod, C, reuse_a, reuse_b)
    v8f d2 = __builtin_amdgcn_wmma_f32_16x16x4_f32(
        false, A2, false, ones, (short)0, c2, false, false);
    v8f d3 = __builtin_amdgcn_wmma_f32_16x16x4_f32(
        false, A3, false, ones, (short)0, c3, false, false);

#pragma unroll
    for (int r = 0; r < 8; r += 2) {
      const float t20 = relu_clamp(d2[r]);
      const float t21 = relu_clamp(d2[r + 1]);
      const float t30 = relu_clamp(d3[r]);
      const float t31 = relu_clamp(d3[r + 1]);
      acc0  = fmaf(t20, t20, acc0);
      acc1  = fmaf(t21, t21, acc1);
      acc2a = fmaf(t30, t30, acc2a);
      acc3a = fmaf(t31, t31, acc3a);
    }
  }

  float acc = (acc0 + acc1) + (acc2a + acc3a);

  // Wave reduction (wave32) — fixed butterfly, deterministic
#pragma unroll
  for (int off = 16; off >= 1; off >>= 1) acc += __shfl_xor(acc, off, 32);

  __shared__ float s_wave[8];
  if (lane == 0) s_wave[wave] = acc;
  __syncthreads();
  if (tid == 0) {
    float s = 0.0f;
#pragma unroll
    for (int w = 0; w < 8; ++w) s += s_wave[w];
    partials[c * 64 + jb] = s;   // exactly one writer per slot
  }
}

// ---------------------------------------------------------------------------
// Kernel 3: fixed-order final reduction + degenerate/NaN handling + mean.
// Single block of 32 threads.
// ---------------------------------------------------------------------------
__global__ void rocstar_final_kernel(const float* __restrict__ stats,
                                     const float* __restrict__ partials,
                                     float* __restrict__ out) {
  __shared__ float res[CC];
  const int t = threadIdx.x;
  if (t < CC) {
    float m = 0.0f;
    for (int jb = 0; jb < 64; ++jb) m += partials[t * 64 + jb];
    float r = m * (1.0f / MAXPOS);       // m2/1000 + m3/1000 combined
    if (isnan(r)) r = 0.0f;
    if (stats[t * 4 + 1] != 0.0f) r = stats[t * 4 + 2] * 1e-8f;
    res[t] = r;
  }
  __syncthreads();
  if (t == 0) {
    float s = 0.0f;
    for (int i = 0; i < CC; ++i) s += res[i];
    out[0] = s / (float)CC;
  }
}

// ---------------------------------------------------------------------------
extern "C" void kernel_launch(void* const* d_in, const int* in_sizes, int n_in,
                              void* d_out, int out_size, void* d_ws, size_t ws_size,
                              hipStream_t stream) {
  const float* y_pred     = (const float*)d_in[0];
  const float* y_true     = (const float*)d_in[1];
  const float* epoch_pred = (const float*)d_in[2];
  const float* epoch_true = (const float*)d_in[3];
  const float* gamma      = (const float*)d_in[4];
  const float* rand_pos   = (const float*)d_in[5];
  const float* rand_neg   = (const float*)d_in[6];
  float* out = (float*)d_out;

  float* stats    = (float*)d_ws;        // CC*4 floats
  float* partials = stats + 80;          // CC*64 floats (fully overwritten)

  rocstar_stats_kernel<<<dim3(CC), dim3(BB), 0, stream>>>(
      y_pred, y_true, epoch_true, stats);

  rocstar_pair_kernel<<<dim3(64, CC), dim3(256), 0, stream>>>(
      y_pred, y_true, epoch_pred, epoch_true, gamma, rand_pos, rand_neg,
      stats, partials);

  rocstar_final_kernel<<<dim3(1), dim3(32), 0, stream>>>(stats, partials, out);
}